// QCNN_Shared_Old_12756052869704
// MI455X (gfx1250) — compile-verified
//
#include <hip/hip_runtime.h>
#include <hip/hip_bf16.h>
#include <math.h>

typedef __attribute__((ext_vector_type(2))) float v2f;
typedef __attribute__((ext_vector_type(8))) float v8f;

// D = A(16x4 f32) * B(4x16 f32) + C(16x16 f32)
__device__ inline v8f wmma4(v2f a, v2f b, v8f c) {
    return __builtin_amdgcn_wmma_f32_16x16x4_f32(
        /*neg_a=*/false, a, /*neg_b=*/false, b,
        /*c_mod=*/(short)0, c, /*reuse_a=*/false, /*reuse_b=*/false);
}

// ---------------- Kernel 1: fold each branch circuit into a 256x256 unitary ----
// u[8] = {u00r,u00i, u01r,u01i, u10r,u10i, u11r,u11i}

__device__ inline void rot_mat(int branch, float theta, float u[8]) {
    float h = 0.5f * theta;
    float c = cosf(h), s = sinf(h);
    if (branch == 0) {        // RX: [[c, -i s],[-i s, c]]
        u[0]=c; u[1]=0.f; u[2]=0.f; u[3]=-s; u[4]=0.f; u[5]=-s; u[6]=c; u[7]=0.f;
    } else if (branch == 1) { // RY: [[c, -s],[s, c]]
        u[0]=c; u[1]=0.f; u[2]=-s; u[3]=0.f; u[4]=s; u[5]=0.f; u[6]=c; u[7]=0.f;
    } else {                  // RZ: diag(e^{-ih}, e^{ih})
        u[0]=c; u[1]=-s; u[2]=0.f; u[3]=0.f; u[4]=0.f; u[5]=0.f; u[6]=c; u[7]=s;
    }
}

__device__ inline void u3_mat(float th, float ph, float la, float u[8]) {
    float c = cosf(0.5f * th), s = sinf(0.5f * th);
    u[0] = c;              u[1] = 0.f;
    u[2] = -cosf(la) * s;  u[3] = -sinf(la) * s;
    u[4] =  cosf(ph) * s;  u[5] =  sinf(ph) * s;
    float cl = cosf(ph + la), sl = sinf(ph + la);
    u[6] = cl * c;         u[7] = sl * c;
}

__device__ inline void apply1q(float* sre, float* sim, int t, int p, const float u[8]) {
    const int bit = 1 << p;
    for (int idx = 0; idx < 256; ++idx) {
        if (idx & bit) continue;
        int i0 = idx * 32 + t, i1 = (idx | bit) * 32 + t;
        float a0r = sre[i0], a0i = sim[i0], a1r = sre[i1], a1i = sim[i1];
        float n0r = u[0]*a0r - u[1]*a0i + u[2]*a1r - u[3]*a1i;
        float n0i = u[0]*a0i + u[1]*a0r + u[2]*a1i + u[3]*a1r;
        float n1r = u[4]*a0r - u[5]*a0i + u[6]*a1r - u[7]*a1i;
        float n1i = u[4]*a0i + u[5]*a0r + u[6]*a1i + u[7]*a1r;
        sre[i0]=n0r; sim[i0]=n0i; sre[i1]=n1r; sim[i1]=n1i;
    }
}

__device__ inline void applyCtrl(float* sre, float* sim, int t, int pc, int pt, const float u[8]) {
    const int cb = 1 << pc, tb = 1 << pt;
    for (int idx = 0; idx < 256; ++idx) {
        if (!(idx & cb) || (idx & tb)) continue;
        int i0 = idx * 32 + t, i1 = (idx | tb) * 32 + t;
        float a0r = sre[i0], a0i = sim[i0], a1r = sre[i1], a1i = sim[i1];
        float n0r = u[0]*a0r - u[1]*a0i + u[2]*a1r - u[3]*a1i;
        float n0i = u[0]*a0i + u[1]*a0r + u[2]*a1i + u[3]*a1r;
        float n1r = u[4]*a0r - u[5]*a0i + u[6]*a1r - u[7]*a1i;
        float n1i = u[4]*a0i + u[5]*a0r + u[6]*a1i + u[7]*a1r;
        sre[i0]=n0r; sim[i0]=n0i; sre[i1]=n1r; sim[i1]=n1i;
    }
}

// grid = (8 column-chunks, 3 branches), block = 32 threads (1 wave).
// Thread t evolves basis column j = blockIdx.x*32 + t privately in LDS, then
// writes W row j (W[k][n] = U[n][k]) directly in the packed WMMA-B layout:
//   float4 Wp[branch][(k>>1)][n] = {Wre[kb][n], Wre[kb+1][n], Wim[kb][n], Wim[kb+1][n]}
// where kb = (k>>1)*2. Thread j contributes component (j&1) / 2+(j&1).
__global__ __launch_bounds__(32) void build_unitaries(
    const float* __restrict__ crx, const float* __restrict__ u3p,
    float* __restrict__ Wp) {
    __shared__ float sre[256 * 32];
    __shared__ float sim[256 * 32];
    const int t = threadIdx.x;
    const int branch = blockIdx.y;
    const int col = blockIdx.x * 32 + t;

    for (int i = 0; i < 256; ++i) {
        sre[i * 32 + t] = (i == col) ? 1.f : 0.f;
        sim[i * 32 + t] = 0.f;
    }

    // wire w lives at bit (7 - w) of the amplitude index
    const int P1c[7] = {0,2,4,6,1,3,5}, P1t[7] = {1,3,5,7,2,4,6};
    const int P2c[3] = {1,5,3},         P2t[3] = {3,7,5};
    const int U3W[4] = {1,3,5,7};
    float u[8];

    for (int i = 0; i < 7; ++i) {
        rot_mat(branch, crx[i], u);
        applyCtrl(sre, sim, t, 7 - P1c[i], 7 - P1t[i], u);
    }
    for (int i = 0; i < 4; ++i) {
        const float* p = u3p + branch * 18 + i * 3;
        u3_mat(p[0], p[1], p[2], u);
        apply1q(sre, sim, t, 7 - U3W[i], u);
    }
    for (int i = 0; i < 3; ++i) {
        rot_mat(branch, crx[7 + i], u);
        applyCtrl(sre, sim, t, 7 - P2c[i], 7 - P2t[i], u);
    }
    {
        const float* p = u3p + branch * 18 + 4 * 3;
        u3_mat(p[0], p[1], p[2], u);
        apply1q(sre, sim, t, 7 - 3, u);
        p = u3p + branch * 18 + 5 * 3;
        u3_mat(p[0], p[1], p[2], u);
        apply1q(sre, sim, t, 7 - 7, u);
    }

    // Packed write: float4 index f = (col>>1)*256 + n; component (col&1)+{0,2}
    float* wb = Wp + branch * 131072;            // 64*2*256 float4 = 131072 floats
    const int comp = col & 1;
    const int base = ((col >> 1) * 256) * 4 + comp;
    for (int n = 0; n < 256; ++n) {
        wb[base + n * 4]     = sre[n * 32 + t];  // Re component
        wb[base + n * 4 + 2] = sim[n * 32 + t];  // Im component
    }
}

// ---------------- Kernel 2: complex GEMM phi = psi * W via WMMA + fused reduce -
// 256 threads = 8 waves; block covers 32 batch rows x all 256 output columns.
// Each wave: 2 M-subtiles x 2 N-tiles (B fragment reused across M subtiles).
#define LDA 260  // stride padding: bank-conflict-free A-fragment reads

__global__ __launch_bounds__(256) void branch_gemm(
    const float* __restrict__ psi_re, const float* __restrict__ psi_im,
    const float* __restrict__ Wp,
    float* __restrict__ feats) {
    __shared__ float are_s[32 * LDA];
    __shared__ float aim_s[32 * LDA];
    __shared__ float facc[32][2];

    const int tid = threadIdx.x;
    const int bm = blockIdx.x;       // 32-row batch tile
    const int branch = blockIdx.y;   // 0..2
    const float4* wp = (const float4*)(Wp + branch * 131072);

    if (tid < 64) facc[tid >> 1][tid & 1] = 0.f;

    for (int idx = tid; idx < 32 * 256; idx += 256) {
        int r = idx >> 8, k = idx & 255;
        are_s[r * LDA + k] = psi_re[(bm * 32 + r) * 256 + k];
        aim_s[r * LDA + k] = psi_im[(bm * 32 + r) * 256 + k];
    }
    __syncthreads();

    const int lane  = tid & 31;
    const int wv    = tid >> 5;          // 8 waves, each owns 32 columns
    const int m15   = lane & 15;         // A-fragment row (ISA 16x4 f32 layout)
    const int khalf = lane >> 4;         // K sub-pair {0,1} -> kk = 2*khalf
    const int kk    = khalf << 1;
    const int nl    = lane & 15;         // B/C column within tile
    const int nbase = wv * 32;

    v8f crr[2][2], cii[2][2], cri[2][2], cir[2][2];  // [msub][ntile]
    {
        v8f z = {};
#pragma unroll
        for (int ms = 0; ms < 2; ++ms)
#pragma unroll
            for (int t2 = 0; t2 < 2; ++t2) {
                crr[ms][t2]=z; cii[ms][t2]=z; cri[ms][t2]=z; cir[ms][t2]=z;
            }
    }

    for (int k = 0; k < 256; k += 4) {
        v2f ar[2], ai[2];
#pragma unroll
        for (int ms = 0; ms < 2; ++ms) {
            int row = (ms * 16 + m15) * LDA + k + kk;
            ar[ms].x = are_s[row]; ar[ms].y = are_s[row + 1];
            ai[ms].x = aim_s[row]; ai[ms].y = aim_s[row + 1];
        }
#pragma unroll
        for (int t2 = 0; t2 < 2; ++t2) {
            int n = nbase + t2 * 16 + nl;
            // packed B fragment: one b128 load gives both K values, Re + Im
            float4 w = wp[((k >> 1) + khalf) * 256 + n];
            v2f br, bi;
            br.x = w.x; br.y = w.y;
            bi.x = w.z; bi.y = w.w;
#pragma unroll
            for (int ms = 0; ms < 2; ++ms) {
                crr[ms][t2] = wmma4(ar[ms], br, crr[ms][t2]);  // Re*Re
                cii[ms][t2] = wmma4(ai[ms], bi, cii[ms][t2]);  // Im*Im
                cri[ms][t2] = wmma4(ar[ms], bi, cri[ms][t2]);  // Re*Im
                cir[ms][t2] = wmma4(ai[ms], br, cir[ms][t2]);  // Im*Re
            }
        }
    }

    // signed |phi|^2 reduction; C layout: VGPR r -> M = r + 8*(lane>>4), N = lane&15
    float part3[2][8], part7[2][8];
#pragma unroll
    for (int ms = 0; ms < 2; ++ms)
#pragma unroll
        for (int r = 0; r < 8; ++r) { part3[ms][r] = 0.f; part7[ms][r] = 0.f; }

#pragma unroll
    for (int t2 = 0; t2 < 2; ++t2) {
        int n = nbase + t2 * 16 + nl;
        float s3 = (n & 16) ? -1.f : 1.f;  // wire 3 -> bit 4
        float s7 = (n & 1)  ? -1.f : 1.f;  // wire 7 -> bit 0
#pragma unroll
        for (int ms = 0; ms < 2; ++ms)
#pragma unroll
            for (int r = 0; r < 8; ++r) {
                float re = crr[ms][t2][r] - cii[ms][t2][r];
                float im = cri[ms][t2][r] + cir[ms][t2][r];
                float p = re * re + im * im;
                part3[ms][r] += s3 * p;
                part7[ms][r] += s7 * p;
            }
    }
    // reduce over the 16 N-lanes (xor of bits 0..3 keeps the M-half bit fixed)
#pragma unroll
    for (int off = 8; off >= 1; off >>= 1) {
#pragma unroll
        for (int ms = 0; ms < 2; ++ms)
#pragma unroll
            for (int r = 0; r < 8; ++r) {
                part3[ms][r] += __shfl_xor(part3[ms][r], off, 32);
                part7[ms][r] += __shfl_xor(part7[ms][r], off, 32);
            }
    }
    if ((lane & 15) == 0) {
        int hi = lane >> 4;
#pragma unroll
        for (int ms = 0; ms < 2; ++ms)
#pragma unroll
            for (int r = 0; r < 8; ++r) {
                int mg = ms * 16 + r + 8 * hi;
                atomicAdd(&facc[mg][0], part3[ms][r]);
                atomicAdd(&facc[mg][1], part7[ms][r]);
            }
    }
    __syncthreads();
    if (tid < 64) {
        int mm = tid >> 1, j = tid & 1;
        feats[(bm * 32 + mm) * 6 + branch * 2 + j] = facc[mm][j];
    }
}

// ---------------- Kernel 3: tiny MLP head ------------------------------------
__global__ __launch_bounds__(256) void mlp_head(
    const float* __restrict__ feats,
    const float* __restrict__ w1, const float* __restrict__ b1,
    const float* __restrict__ w2, const float* __restrict__ b2,
    float* __restrict__ out, int B) {
    int b = blockIdx.x * blockDim.x + threadIdx.x;
    if (b >= B) return;
    float f[6];
#pragma unroll
    for (int i = 0; i < 6; ++i) f[i] = feats[b * 6 + i];
    float o = b2[0];
#pragma unroll
    for (int j = 0; j < 15; ++j) {
        float h = b1[j];
#pragma unroll
        for (int i = 0; i < 6; ++i) h += f[i] * w1[j * 6 + i];
        o += tanhf(h) * w2[j];
    }
    out[b] = 1.f / (1.f + expf(-o));
}

// ---------------- Launch ------------------------------------------------------
extern "C" void kernel_launch(void* const* d_in, const int* in_sizes, int n_in,
                              void* d_out, int out_size, void* d_ws, size_t ws_size,
                              hipStream_t stream) {
    const float* state_re = (const float*)d_in[0];
    const float* state_im = (const float*)d_in[1];
    const float* crx      = (const float*)d_in[2];
    const float* u3p      = (const float*)d_in[3];
    const float* w1       = (const float*)d_in[4];
    const float* b1       = (const float*)d_in[5];
    const float* w2       = (const float*)d_in[6];
    const float* b2       = (const float*)d_in[7];

    float* ws    = (float*)d_ws;
    float* Wp    = ws;                  // 3 * 131072 floats (packed B fragments)
    float* feats = ws + 3 * 131072;     // B * 6 floats

    const int B = in_sizes[0] / 256;

    build_unitaries<<<dim3(8, 3), 32, 0, stream>>>(crx, u3p, Wp);
    branch_gemm<<<dim3(B / 32, 3), 256, 0, stream>>>(state_re, state_im, Wp, feats);
    mlp_head<<<(B + 255) / 256, 256, 0, stream>>>(feats, w1, b1, w2, b2, (float*)d_out, B);
}